// MultiBoxLoss_combined_1726576857236
// MI455X (gfx1250) — compile-verified
//
#include <hip/hip_runtime.h>
#include <stdint.h>

#define BATCH  16
#define PRI    32768
#define OBJ_N  64
#define NCL    80          // NUM_CLASSES-1
#define THRESH 0.5f
#define NEGPOS 3
#define NBLK   2048        // BATCH * PRI/256
#define K4MAX  4096        // BATCH * PRI/128

// ---------------- workspace layout (bytes) ----------------
#define OFF_BPK   0u                                   // u64[B][O] best-prior keys
#define OFF_BTO   (OFF_BPK + 8u*BATCH*OBJ_N)           // f32[B][P] best truth overlap
#define OFF_BTI   (OFF_BTO + 4u*BATCH*PRI)             // i32[B][P] best truth idx
#define OFF_LO    (OFF_BTI + 4u*BATCH*PRI)             // f32[B][P] objectness CE (rank keys)
#define OFF_BLL   (OFF_LO  + 4u*BATCH*PRI)             // f32[NBLK] loss_l block partials
#define OFF_BOP   (OFF_BLL + 4u*NBLK)                  // f32[NBLK] loss_obj (pos) partials
#define OFF_BNP   (OFF_BOP + 4u*NBLK)                  // f32[NBLK] num_pos partials
#define OFF_BPC   (OFF_BNP + 4u*NBLK)                  // i32[NBLK] pos counts
#define OFF_BPO   (OFF_BPC + 4u*NBLK)                  // i32[NBLK] pos offsets (scan)
#define OFF_MISC  (OFF_BPO + 4u*NBLK)                  // 1 KB small state
#define OFF_K4P   (OFF_MISC + 1024u)                   // f32[K4MAX] conf-loss block partials
#define OFF_ENT   (OFF_K4P + 4u*K4MAX)                 // int4[capacity] positive entries

// misc word indices (32-bit words at OFF_MISC)
#define MI_NUMNEG 0     // [b] int
#define MI_THR    16    // [b] uint  (k-th largest lo, bit pattern)
#define MI_REM    32    // [b] int   (#tied elements to include)
#define MI_NEGSUM 48    // [b] float
#define MI_LL     64    // [b] float
#define MI_LOP    80    // [b] float
#define MI_NUMPOS 96    // [b] int
#define MI_TOTAL  112   // int: total positives
#define MI_NPAD   113   // int: positives rounded up to 16 (clamped to capacity)

typedef float v2f __attribute__((ext_vector_type(2)));
typedef float v8f __attribute__((ext_vector_type(8)));

__device__ __forceinline__ float sl1(float d) {
  float a = fabsf(d);
  return (a < 1.0f) ? 0.5f * d * d : a - 0.5f;
}

// ---------------- K0: init best-prior keys ----------------
__global__ void k0_init(void* ws) {
  int i = blockIdx.x * 256 + threadIdx.x;
  unsigned long long* bpk = (unsigned long long*)((char*)ws + OFF_BPK);
  if (i < BATCH * OBJ_N) bpk[i] = 0ull;
}

// ---------------- K1: IoU matching ----------------
__global__ void k1_match(const float* __restrict__ priors,
                         const float* __restrict__ targets, void* ws) {
  const int blk = blockIdx.x;
  const int b = blk >> 7;
  const int p = ((blk & 127) << 8) + threadIdx.x;
  float* bto = (float*)((char*)ws + OFF_BTO) + b * PRI;
  int*   bti = (int*)((char*)ws + OFF_BTI) + b * PRI;
  unsigned long long* bpk = (unsigned long long*)((char*)ws + OFF_BPK) + b * OBJ_N;

  __shared__ float st[OBJ_N * 6];
  __shared__ unsigned long long skey[OBJ_N];
  for (int i = threadIdx.x; i < OBJ_N * 6; i += 256) st[i] = targets[(size_t)b * OBJ_N * 6 + i];
  if (threadIdx.x < OBJ_N) skey[threadIdx.x] = 0ull;
  __syncthreads();

  float4 pr = ((const float4*)priors)[p];
  float px0 = pr.x - 0.5f * pr.z, py0 = pr.y - 0.5f * pr.w;
  float px1 = pr.x + 0.5f * pr.z, py1 = pr.y + 0.5f * pr.w;
  float area_p = (px1 - px0) * (py1 - py0);
  float best = -1.0f; int bidx = 0;
  for (int t = 0; t < OBJ_N; ++t) {
    float tx0 = st[t*6+0], ty0 = st[t*6+1], tx1 = st[t*6+2], ty1 = st[t*6+3];
    float iw = fmaxf(fminf(tx1, px1) - fmaxf(tx0, px0), 0.f);
    float ih = fmaxf(fminf(ty1, py1) - fmaxf(ty0, py0), 0.f);
    float inter = iw * ih;
    float area_t = (tx1 - tx0) * (ty1 - ty0);
    float iou = inter / (area_t + area_p - inter);
    if (iou > best) { best = iou; bidx = t; }           // first-index tie-break
    unsigned long long key = ((unsigned long long)__float_as_uint(iou) << 32)
                           | (unsigned)(PRI - 1 - p);   // smallest p wins ties
    atomicMax(&skey[t], key);
  }
  bto[p] = best; bti[p] = bidx;
  __syncthreads();
  if (threadIdx.x < OBJ_N) atomicMax(&bpk[threadIdx.x], skey[threadIdx.x]);
}

// ---------------- K2: force-match override (serial per batch) ----------------
__global__ void k2_override(void* ws) {
  int b = threadIdx.x;
  if (b >= BATCH) return;
  unsigned long long* bpk = (unsigned long long*)((char*)ws + OFF_BPK) + b * OBJ_N;
  float* bto = (float*)((char*)ws + OFF_BTO) + b * PRI;
  int*   bti = (int*)((char*)ws + OFF_BTI) + b * PRI;
  for (int t = 0; t < OBJ_N; ++t) {
    int p = PRI - 1 - (int)(unsigned)(bpk[t] & 0xFFFFFFFFull);
    bto[p] = 2.0f; bti[p] = t;
  }
}

// ---------------- K3: loc loss + objectness CE + pos stats ----------------
__global__ void k3_main(const float* __restrict__ loc, const float* __restrict__ obj,
                        const float* __restrict__ priors, const float* __restrict__ targets,
                        void* ws) {
  const int blk = blockIdx.x;
  const int b = blk >> 7;
  const int tid = threadIdx.x;
  const int p = ((blk & 127) << 8) + tid;
  const int gp = b * PRI + p;
  const float* bto = (const float*)((char*)ws + OFF_BTO);
  const int*   bti = (const int*)((char*)ws + OFF_BTI);
  float* lo_arr = (float*)((char*)ws + OFF_LO);

  float ov = bto[gp];
  int t = bti[gp];
  const float* tr = targets + ((size_t)b * OBJ_N + t) * 6;
  float m0 = tr[0], m1 = tr[1], M0 = tr[2], M1 = tr[3], w = tr[5];
  bool pos = ov >= THRESH;
  float wgt = pos ? w : 1.0f;

  float4 pr = ((const float4*)priors)[p];
  float gx = ((m0 + M0) * 0.5f - pr.x) / (0.1f * pr.z);
  float gy = ((m1 + M1) * 0.5f - pr.y) / (0.1f * pr.w);
  float gw = logf((M0 - m0) / pr.z) / 0.2f;
  float gh = logf((M1 - m1) / pr.w) / 0.2f;
  float4 ld = ((const float4*)loc)[gp];
  float s = sl1(ld.x - gx) + sl1(ld.y - gy) + sl1(ld.z - gw) + sl1(ld.w - gh);
  float c_l = pos ? s * wgt : 0.f;

  float2 ob = ((const float2*)obj)[gp];
  float m = fmaxf(ob.x, ob.y);
  float lse = m + logf(expf(ob.x - m) + expf(ob.y - m));
  lo_arr[gp] = pos ? 0.f : (lse - ob.x);
  float c_o = pos ? (lse - ob.y) * wgt : 0.f;
  float c_n = pos ? wgt : 0.f;

  __shared__ float s0[256], s1[256], s2[256];
  __shared__ int si[256];
  s0[tid] = c_l; s1[tid] = c_o; s2[tid] = c_n; si[tid] = pos ? 1 : 0;
  __syncthreads();
  for (int st = 128; st > 0; st >>= 1) {
    if (tid < st) { s0[tid]+=s0[tid+st]; s1[tid]+=s1[tid+st]; s2[tid]+=s2[tid+st]; si[tid]+=si[tid+st]; }
    __syncthreads();
  }
  if (tid == 0) {
    ((float*)((char*)ws + OFF_BLL))[blk] = s0[0];
    ((float*)((char*)ws + OFF_BOP))[blk] = s1[0];
    ((float*)((char*)ws + OFF_BNP))[blk] = s2[0];
    ((int*)((char*)ws + OFF_BPC))[blk] = si[0];
  }
}

// ---------------- K5: global scan of positive counts ----------------
__global__ void k5_scan(void* ws, int capacity) {
  const int tid = threadIdx.x;
  const int* pc = (const int*)((char*)ws + OFF_BPC);
  int* po = (int*)((char*)ws + OFF_BPO);
  __shared__ int part[256];
  __shared__ int excl[257];
  int local[8]; int sum = 0;
  for (int i = 0; i < 8; ++i) { local[i] = pc[tid * 8 + i]; sum += local[i]; }
  part[tid] = sum; __syncthreads();
  if (tid == 0) { int acc = 0; for (int i = 0; i < 256; ++i) { excl[i] = acc; acc += part[i]; } excl[256] = acc; }
  __syncthreads();
  int off = excl[tid];
  for (int i = 0; i < 8; ++i) { po[tid * 8 + i] = off; off += local[i]; }
  if (tid == 0) {
    int total = excl[256];
    int npad = (total + 15) & ~15;
    if (npad > capacity) npad = capacity;
    if (total > capacity) total = capacity;
    int* mi = (int*)((char*)ws + OFF_MISC);
    mi[MI_TOTAL] = total; mi[MI_NPAD] = npad;
  }
}

// ---------------- K8: per-batch sums (loss_l, loss_obj_pos, num_pos, num_neg) ----------------
__global__ void k8_batch(void* ws) {
  int b = blockIdx.x, tid = threadIdx.x;
  const float* bll = (const float*)((char*)ws + OFF_BLL);
  const float* bop = (const float*)((char*)ws + OFF_BOP);
  const float* bnp = (const float*)((char*)ws + OFF_BNP);
  __shared__ float a0[128], a1[128], a2[128];
  a0[tid] = bll[b*128+tid]; a1[tid] = bop[b*128+tid]; a2[tid] = bnp[b*128+tid];
  __syncthreads();
  for (int st = 64; st > 0; st >>= 1) {
    if (tid < st) { a0[tid]+=a0[tid+st]; a1[tid]+=a1[tid+st]; a2[tid]+=a2[tid+st]; }
    __syncthreads();
  }
  if (tid == 0) {
    float* mf = (float*)((char*)ws + OFF_MISC);
    int*   mi = (int*)((char*)ws + OFF_MISC);
    mf[MI_LL + b] = a0[0]; mf[MI_LOP + b] = a1[0];
    int np = (int)a2[0];
    mi[MI_NUMPOS + b] = np;
    int nn = np * NEGPOS; if (nn > PRI - 1) nn = PRI - 1;
    mi[MI_NUMNEG + b] = nn;
  }
}

// ---------------- K6: compact positives into dense entry list ----------------
__global__ void k6_fill(const float* __restrict__ obj, const float* __restrict__ targets,
                        const float* __restrict__ conf, void* ws, int capacity) {
  const int blk = blockIdx.x;
  const int b = blk >> 7;
  const int tid = threadIdx.x;
  const int p = ((blk & 127) << 8) + tid;
  const int gp = b * PRI + p;
  const float* bto = (const float*)((char*)ws + OFF_BTO);
  const int*   bti = (const int*)((char*)ws + OFF_BTI);
  bool pos = bto[gp] >= THRESH;
  __shared__ int sflag[256];
  __shared__ int excl[256];
  sflag[tid] = pos ? 1 : 0;
  __syncthreads();
  if (tid == 0) { int acc = 0; for (int i = 0; i < 256; ++i) { excl[i] = acc; acc += sflag[i]; } }
  __syncthreads();
  if (pos) {
    int idx = ((const int*)((char*)ws + OFF_BPO))[blk] + excl[tid];
    if (idx < capacity) {
      int t = bti[gp];
      const float* tr = targets + ((size_t)b * OBJ_N + t) * 6;
      int k = (int)tr[4]; k = (k < 1) ? 1 : (k > NCL ? NCL : k);
      float2 ob = ((const float2*)obj)[gp];
      float m = fmaxf(ob.x, ob.y);
      float lse = m + logf(expf(ob.x - m) + expf(ob.y - m));
      int4 e; e.x = gp; e.y = k;
      e.z = __float_as_int(lse - ob.y);  // lse_obj - obj1
      e.w = __float_as_int(tr[5]);       // weight
      ((int4*)((char*)ws + OFF_ENT))[idx] = e;
      __builtin_prefetch(conf + (size_t)gp * NCL, 0, 1);  // pull gathered conf row toward cache
    }
  }
}

// ---------------- K7: pad entry list to multiple of 16 ----------------
__global__ void k7_pad(void* ws) {
  const int* mi = (const int*)((char*)ws + OFF_MISC);
  int i = mi[MI_TOTAL] + threadIdx.x;
  if (i < mi[MI_NPAD]) {
    int4 e; e.x = 0; e.y = 1; e.z = __float_as_int(0.f); e.w = __float_as_int(0.f);
    ((int4*)((char*)ws + OFF_ENT))[i] = e;
  }
}

// ---------------- K9: per-batch radix select of k-th largest lo ----------------
__global__ void k9_radix(void* ws) {
  int b = blockIdx.x, tid = threadIdx.x;
  int* mi = (int*)((char*)ws + OFF_MISC);
  unsigned* mu = (unsigned*)mi;
  int k = mi[MI_NUMNEG + b];
  const unsigned* lob = (const unsigned*)((char*)ws + OFF_LO) + (size_t)b * PRI;
  if (k <= 0) { if (tid == 0) { mu[MI_THR + b] = 0xFFFFFFFFu; mi[MI_REM + b] = 0; } return; }
  __shared__ unsigned hist[256];
  __shared__ unsigned s_prefix;
  __shared__ int s_kk;
  if (tid == 0) { s_prefix = 0u; s_kk = k; }
  __syncthreads();
  for (int round = 0; round < 4; ++round) {
    hist[tid] = 0;
    __syncthreads();
    const int shift = 24 - 8 * round;
    const unsigned mask = (round == 0) ? 0u : (0xFFFFFFFFu << (32 - 8 * round));
    const unsigned pref = s_prefix;
    for (int i = tid; i < PRI; i += 256) {
      unsigned v = lob[i];
      if ((v & mask) == pref) atomicAdd(&hist[(v >> shift) & 255], 1u);
    }
    __syncthreads();
    if (tid == 0) {
      unsigned kk = (unsigned)s_kk, cum = 0;
      for (int bin = 255; bin >= 0; --bin) {
        unsigned c = hist[bin];
        if (cum + c >= kk) { s_kk = (int)(kk - cum); s_prefix = pref | ((unsigned)bin << shift); break; }
        cum += c;
      }
    }
    __syncthreads();
  }
  if (tid == 0) { mu[MI_THR + b] = s_prefix; mi[MI_REM + b] = s_kk; }
}

// ---------------- K10: per-batch sum of selected negative CE ----------------
__global__ void k10_negsum(void* ws) {
  int b = blockIdx.x, tid = threadIdx.x;
  int* mi = (int*)((char*)ws + OFF_MISC);
  unsigned* mu = (unsigned*)mi;
  float* mf = (float*)mi;
  unsigned thr = mu[MI_THR + b];
  int rem = mi[MI_REM + b];
  const unsigned* lob = (const unsigned*)((char*)ws + OFF_LO) + (size_t)b * PRI;
  float acc = 0.f;
  for (int i = tid; i < PRI; i += 256) {
    unsigned v = lob[i];
    if (v > thr) acc += __uint_as_float(v);
  }
  __shared__ float sm[256];
  sm[tid] = acc; __syncthreads();
  for (int st = 128; st > 0; st >>= 1) { if (tid < st) sm[tid] += sm[tid + st]; __syncthreads(); }
  if (tid == 0) {
    float tot = sm[0];
    if (rem > 0) tot += (float)rem * __uint_as_float(thr);  // tied elements contribute identically
    mf[MI_NEGSUM + b] = tot;
  }
}

// ---------------- K4: conf loss over positives via WMMA exp-row-sums ----------------
__global__ void k4_conf(const float* __restrict__ conf, void* ws) {
  const int tid = threadIdx.x;
  const int wave = tid >> 5, lane = tid & 31;
  const int l16 = lane & 15, hi = lane >> 4;
  const int* mi = (const int*)((char*)ws + OFF_MISC);
  const int npad = mi[MI_NPAD];
  const int base = blockIdx.x * 128 + wave * 16;    // 16 entries per wave
  const int4* ent = (const int4*)((char*)ws + OFF_ENT);
  __shared__ float smS[128];   // 8 waves * 16 row-sums
  __shared__ float smR[256];
  float term = 0.f;
  const bool active = base < npad;                   // wave-uniform: EXEC all-ones for WMMA
  if (active) {
    int4 e = ent[base + l16];
    const long rowoff = (long)e.x * NCL;
    v8f acc = {0.f,0.f,0.f,0.f,0.f,0.f,0.f,0.f};
    v2f ones; ones[0] = 1.f; ones[1] = 1.f;
    // A(16x4) layout: lanes0-15 hold K=c0..c0+1, lanes16-31 hold K=c0+2..c0+3; B = ones -> row sums
    for (int c0 = 0; c0 < NCL; c0 += 4) {
      float x0 = conf[rowoff + c0 + 2 * hi];
      float x1 = conf[rowoff + c0 + 2 * hi + 1];
      v2f a; a[0] = expf(x0); a[1] = expf(x1);
      acc = __builtin_amdgcn_wmma_f32_16x16x4_f32(false, a, false, ones, (short)0, acc,
                                                  false, false);
    }
    // D col N=0: lane0 holds rows 0-7 (vgpr j), lane16 holds rows 8-15
    if (lane == 0)  { for (int j = 0; j < 8; ++j) smS[wave * 16 + j]     = acc[j]; }
    if (lane == 16) { for (int j = 0; j < 8; ++j) smS[wave * 16 + 8 + j] = acc[j]; }
  }
  __syncthreads();
  if (active && hi == 0) {
    int4 e = ent[base + l16];
    float S = smS[wave * 16 + l16];
    float basev = __int_as_float(e.z);   // lse_obj - obj1
    float wq = __int_as_float(e.w);
    float ck = conf[(long)e.x * NCL + (e.y - 1)];
    term = (logf(S) + basev - ck) * wq;  // lse_conf + lse_obj - obj1 - conf[cls-1]
  }
  smR[tid] = term; __syncthreads();
  for (int st = 128; st > 0; st >>= 1) { if (tid < st) smR[tid] += smR[tid + st]; __syncthreads(); }
  if (tid == 0) ((float*)((char*)ws + OFF_K4P))[blockIdx.x] = smR[0];
}

// ---------------- K11: final combine ----------------
__global__ void k11_final(void* ws, float* out, int nblk4) {
  int tid = threadIdx.x;
  const float* k4p = (const float*)((char*)ws + OFF_K4P);
  float acc = 0.f;
  for (int i = tid; i < nblk4; i += 256) acc += k4p[i];
  __shared__ float sm[256];
  sm[tid] = acc; __syncthreads();
  for (int st = 128; st > 0; st >>= 1) { if (tid < st) sm[tid] += sm[tid + st]; __syncthreads(); }
  if (tid == 0) {
    float* mf = (float*)((char*)ws + OFF_MISC);
    int*   mi = (int*)((char*)ws + OFF_MISC);
    float Ll = 0.f, Lo = 0.f; int np = 0;
    for (int b = 0; b < BATCH; ++b) {
      Ll += mf[MI_LL + b];
      Lo += mf[MI_LOP + b] + mf[MI_NEGSUM + b];
      np += mi[MI_NUMPOS + b];
    }
    float N = (float)(np > 0 ? np : 1);
    out[0] = Ll / N; out[1] = sm[0] / N; out[2] = Lo / N;
  }
}

// ---------------- host launcher ----------------
extern "C" void kernel_launch(void* const* d_in, const int* in_sizes, int n_in,
                              void* d_out, int out_size, void* d_ws, size_t ws_size,
                              hipStream_t stream) {
  const float* loc    = (const float*)d_in[0];
  const float* conf   = (const float*)d_in[1];
  const float* obj    = (const float*)d_in[2];
  const float* priors = (const float*)d_in[3];
  const float* tgts   = (const float*)d_in[4];
  float* out = (float*)d_out;

  long long avail = (long long)ws_size - (long long)OFF_ENT;
  int capacity = BATCH * PRI;
  if (avail < (long long)capacity * 16) capacity = (int)(avail > 0 ? avail / 16 : 0);
  capacity &= ~15;
  if (capacity < 16) capacity = 16;
  int nblk4 = (capacity + 127) / 128;

  k0_init   <<<dim3(4),    dim3(256), 0, stream>>>(d_ws);
  k1_match  <<<dim3(NBLK), dim3(256), 0, stream>>>(priors, tgts, d_ws);
  k2_override<<<dim3(1),   dim3(16),  0, stream>>>(d_ws);
  k3_main   <<<dim3(NBLK), dim3(256), 0, stream>>>(loc, obj, priors, tgts, d_ws);
  k5_scan   <<<dim3(1),    dim3(256), 0, stream>>>(d_ws, capacity);
  k8_batch  <<<dim3(BATCH),dim3(128), 0, stream>>>(d_ws);
  k6_fill   <<<dim3(NBLK), dim3(256), 0, stream>>>(obj, tgts, conf, d_ws, capacity);
  k7_pad    <<<dim3(1),    dim3(16),  0, stream>>>(d_ws);
  k9_radix  <<<dim3(BATCH),dim3(256), 0, stream>>>(d_ws);
  k10_negsum<<<dim3(BATCH),dim3(256), 0, stream>>>(d_ws);
  k4_conf   <<<dim3(nblk4),dim3(256), 0, stream>>>(conf, d_ws);
  k11_final <<<dim3(1),    dim3(256), 0, stream>>>(d_ws, out, nblk4);
}